// CollapsePool_58033598104033
// MI455X (gfx1250) — compile-verified
//
#include <hip/hip_runtime.h>
#include <hip/hip_bf16.h>

// Problem constants (from reference)
#define G_     16
#define M_     512
#define D_     128
#define KDEG_  8
#define KREM_  384          // M - TGT nodes removed per graph
#define KEPT_  128
#define N_     (G_ * M_)            // 8192
#define EHALF_ (G_ * M_ * KDEG_)    // 65536 edges per direction
#define XNEW_SZ (G_ * KEPT_ * D_)   // 262144
#define AK_SZ   (G_ * KEPT_ * KEPT_)// 262144
#define BN_SZ   (G_ * KEPT_)        // 2048
#define TPAD_  (M_ + 2)             // LDS row pad: banks 2r/2r+1 per row -> conflict-free

typedef __attribute__((ext_vector_type(2))) float v2f;
typedef __attribute__((ext_vector_type(8))) float v8f;

// ---------------------------------------------------------------------------
// Kernel 1: mag[n] = sum(x[n,:]^2).  One wave32 per node, float4 loads.
// ---------------------------------------------------------------------------
__global__ void k_mag(const float* __restrict__ x, float* __restrict__ mag) {
  int wave = (blockIdx.x * blockDim.x + threadIdx.x) >> 5;
  int lane = threadIdx.x & 31;
  const float4 v = ((const float4*)(x + (size_t)wave * D_))[lane];
  float s = v.x * v.x + v.y * v.y + v.z * v.z + v.w * v.w;
  #pragma unroll
  for (int off = 16; off; off >>= 1) s += __shfl_xor(s, off);
  if (lane == 0) mag[wave] = s;
}

// ---------------------------------------------------------------------------
// Kernel 2: per-graph stable argsort of mag (512 elems, bitonic on packed
// (magBits<<32)|id keys; mag>=0 so IEEE bits order as unsigned).  Emits
// removal order, sorted kept ids, and kept-inverse map.  1 block / graph.
// ---------------------------------------------------------------------------
__global__ void k_sort(const float* __restrict__ mag, int* __restrict__ order,
                       int* __restrict__ kept, int* __restrict__ keptInv) {
  const int g = blockIdx.x, t = threadIdx.x;   // 512 threads
  __shared__ unsigned long long key[M_];
  __shared__ int kid[KEPT_];
  unsigned mb = __float_as_uint(mag[g * M_ + t]);
  key[t] = ((unsigned long long)mb << 32) | (unsigned)t;
  __syncthreads();
  for (int k = 2; k <= M_; k <<= 1)
    for (int j = k >> 1; j > 0; j >>= 1) {
      int ixj = t ^ j;
      if (ixj > t) {
        bool up = ((t & k) == 0);
        unsigned long long a = key[t], b = key[ixj];
        if ((up && a > b) || (!up && a < b)) { key[t] = b; key[ixj] = a; }
      }
      __syncthreads();
    }
  int id = (int)(key[t] & 0xffffffffu);
  order[g * M_ + t] = id;
  keptInv[g * M_ + t] = -1;
  if (t >= KREM_) kid[t - KREM_] = id;
  __syncthreads();
  // sort the 128 kept ids ascending (reference: kept = sort(order[K:]))
  for (int k = 2; k <= KEPT_; k <<= 1)
    for (int j = k >> 1; j > 0; j >>= 1) {
      if (t < KEPT_) {
        int ixj = t ^ j;
        if (ixj > t) {
          bool up = ((t & k) == 0);
          int a = kid[t], b = kid[ixj];
          if ((up && a > b) || (!up && a < b)) { kid[t] = b; kid[ixj] = a; }
        }
      }
      __syncthreads();
    }
  if (t < KEPT_) {
    kept[g * KEPT_ + t] = kid[t];
    keptInv[g * M_ + kid[t]] = t;
  }
}

// ---------------------------------------------------------------------------
// Kernel 3: sequential greedy contraction.  One 512-thread block per graph.
// Adjacency = 512 x 512-bit rows in LDS (32KB).  Per step: wave0 does the
// min-mag-neighbor argmin (id tie-break) via shfl reduce; all 512 threads
// rewire rows/cols with LDS atomics.  Then pointer-doubling resolves each
// node's final owner (root) and merge depth -> coeff = 2^-depth, which is
// exactly T_kept without ever materializing the 512x512 T.
// ---------------------------------------------------------------------------
__global__ void k_contract(const int* __restrict__ ei0, const int* __restrict__ ei1,
                           const float* __restrict__ mag, const int* __restrict__ order,
                           unsigned* __restrict__ Afin, int* __restrict__ root,
                           float* __restrict__ coeff) {
  const int g = blockIdx.x, t = threadIdx.x;   // 512 threads
  __shared__ unsigned A_s[M_][16];
  __shared__ float mag_s[M_];
  __shared__ int removed_s[KREM_];
  __shared__ int par_s[M_];
  __shared__ int dep_s[M_];
  __shared__ unsigned tmp_row[16];
  __shared__ unsigned oth_row[16];
  __shared__ int sh_new;

  #pragma unroll
  for (int w = 0; w < 16; ++w) A_s[t][w] = 0u;
  mag_s[t] = mag[g * M_ + t];
  par_s[t] = t;
  if (t < KREM_) removed_s[t] = order[g * M_ + t];
  __syncthreads();

  const int ebase = g * (M_ * KDEG_);
  for (int e = t; e < M_ * KDEG_; e += M_) {
    int s0 = ei0[ebase + e] & (M_ - 1), d0 = ei1[ebase + e] & (M_ - 1);
    atomicOr(&A_s[s0][d0 >> 5], 1u << (d0 & 31));
    int s1 = ei0[EHALF_ + ebase + e] & (M_ - 1), d1 = ei1[EHALF_ + ebase + e] & (M_ - 1);
    atomicOr(&A_s[s1][d1 >> 5], 1u << (d1 & 31));
  }
  __syncthreads();

  for (int step = 0; step < KREM_; ++step) {
    const int old = removed_s[step];
    if (t < 16) tmp_row[t] = A_s[old][t];
    __syncthreads();
    if (t < 32) {   // wave 0: argmin over neighbors, (mag, id) lexicographic
      float bm = __builtin_inff();
      int bi = 0x7fffffff;
      for (int c = t * 16, ce = c + 16; c < ce; ++c) {
        if (c != old && ((tmp_row[c >> 5] >> (c & 31)) & 1u)) {
          float m = mag_s[c];
          if (m < bm || (m == bm && c < bi)) { bm = m; bi = c; }
        }
      }
      #pragma unroll
      for (int off = 16; off; off >>= 1) {
        float om = __shfl_xor(bm, off);
        int oi = __shfl_xor(bi, off);
        if (om < bm || (om == bm && oi < bi)) { bm = om; bi = oi; }
      }
      if (t == 0) {
        int nw = (bi == 0x7fffffff) ? old : bi;
        sh_new = nw;
        par_s[old] = nw;
      }
    }
    __syncthreads();
    const int nw = sh_new;
    if (t < 16) {
      unsigned o = tmp_row[t];
      if ((old >> 5) == t) o &= ~(1u << (old & 31));
      if ((nw >> 5) == t) o &= ~(1u << (nw & 31));
      oth_row[t] = o;
    }
    __syncthreads();
    atomicAnd(&A_s[t][old >> 5], ~(1u << (old & 31)));          // clear col old
    if ((oth_row[t >> 5] >> (t & 31)) & 1u)
      atomicOr(&A_s[t][nw >> 5], 1u << (nw & 31));              // col new |= others
    if (t < 16) {
      atomicOr(&A_s[nw][t], oth_row[t]);                        // row new |= others
      A_s[old][t] = 0u;                                         // clear row old
    }
    __syncthreads();
  }

  // pointer doubling: par -> final root, dep -> #merge hops (coeff = 2^-dep)
  dep_s[t] = (par_s[t] == t) ? 0 : 1;
  __syncthreads();
  for (int it = 0; it < 10; ++it) {   // 2^10 > 384 max chain length
    int p = par_s[t];
    int nd = dep_s[t] + dep_s[p];
    int np = par_s[p];
    __syncthreads();
    par_s[t] = np;
    dep_s[t] = nd;
    __syncthreads();
  }
  root[g * M_ + t] = par_s[t];
  coeff[g * M_ + t] = ldexpf(1.0f, -dep_s[t]);
  #pragma unroll
  for (int w = 0; w < 16; ++w) Afin[(g * M_ + t) * 16 + w] = A_s[t][w];
}

// ---------------------------------------------------------------------------
// Kernel 4: x_new = T_kept @ x via V_WMMA_F32_16X16X4_F32 (f32 in, f32 acc —
// bit-exact vs FMA).  One block per (graph, 16-row stripe of T_kept); all 8
// waves share that stripe, so it is materialized ONCE per block as a dense
// 16x512 f32 panel in LDS (branchless fill, padded stride -> conflict-free
// ds_load_b64 A-fragments).  Each wave owns one 16-col tile of x_new and runs
// a 128-deep K-loop of WMMAs.
// A(16x4): lanes 0-15 rows, VGPR0/1 = K0/K1; lanes 16-31 = K2/K3.
// B(4x16): lanes 0-15 cols with K0/K1 in the 2 VGPRs; lanes 16-31 = K2/K3.
// C(16x16): VGPR j -> M=j (lanes 0-15) / M=8+j (lanes 16-31).
// ---------------------------------------------------------------------------
__global__ void k_gemm(const float* __restrict__ x, const int* __restrict__ root,
                       const float* __restrict__ coeff, const int* __restrict__ kept,
                       float* __restrict__ xnew) {
  const int g = blockIdx.x >> 3;      // 8 blocks per graph (one per row-stripe)
  const int bset = blockIdx.x & 7;
  const int t = threadIdx.x;          // 256 threads = 8 waves
  __shared__ float T_s[16][TPAD_];    // dense T_kept stripe, rows tm..tm+15
  __shared__ int kept_s[16];
  const int tm = bset * 16;
  if (t < 16) kept_s[t] = kept[g * KEPT_ + tm + t];
  __syncthreads();
  // branchless stripe fill: 16*512 entries, 32 per thread
  for (int e = t; e < 16 * M_; e += 256) {
    const int r = e >> 9;             // row 0..15
    const int m = e & (M_ - 1);       // col 0..511
    const int rt = root[g * M_ + m];
    const float cv = coeff[g * M_ + m];
    T_s[r][m] = (rt == kept_s[r]) ? cv : 0.0f;
  }
  __syncthreads();

  const int wave = t >> 5, lane = t & 31;
  const int tn = wave * 16;           // this wave's 16-col tile
  const int half = lane >> 4, l15 = lane & 15;
  const float* xp = x + (size_t)g * M_ * D_ + (size_t)(half * 2) * D_ + tn + l15;

  v8f c = {};
  for (int kb = 0; kb < M_; kb += 4) {
    const int m0 = kb + half * 2;     // even -> 8B-aligned ds_load_b64
    v2f a = *(const v2f*)&T_s[l15][m0];
    v2f b;
    b.x = xp[0];
    b.y = xp[D_];
    __builtin_prefetch(xp + 8 * D_, 0, 0);   // global_prefetch_b8 lookahead
    c = __builtin_amdgcn_wmma_f32_16x16x4_f32(
        /*neg_a=*/false, a, /*neg_b=*/false, b,
        /*c_mod=*/(short)0, c, /*reuse_a=*/false, /*reuse_b=*/false);
    xp += 4 * D_;
  }

  float* og = xnew + ((size_t)g * KEPT_ + tm + half * 8) * D_ + tn + l15;
  #pragma unroll
  for (int j = 0; j < 8; ++j) og[(size_t)j * D_] = c[j];
}

// ---------------------------------------------------------------------------
// Kernel 5: remaining outputs — A_kept (gather kept x kept bits from final
// adjacency), batch_new, x_mask.  All emitted as f32 into d_out.
// ---------------------------------------------------------------------------
__global__ void k_outputs(const unsigned* __restrict__ Afin, const int* __restrict__ kept,
                          const int* __restrict__ keptInv, float* __restrict__ out) {
  int idx = blockIdx.x * blockDim.x + threadIdx.x;
  if (idx < AK_SZ) {
    int g = idx >> 14;          // / (128*128)
    int r = idx & 16383;
    int i = r >> 7, j = r & 127;
    int ni = kept[g * KEPT_ + i], nj = kept[g * KEPT_ + j];
    unsigned w = Afin[(g * M_ + ni) * 16 + (nj >> 5)];
    out[XNEW_SZ + idx] = ((w >> (nj & 31)) & 1u) ? 1.0f : 0.0f;
  } else if (idx < AK_SZ + BN_SZ) {
    int r = idx - AK_SZ;
    out[XNEW_SZ + AK_SZ + r] = (float)(r >> 7);
  } else if (idx < AK_SZ + BN_SZ + N_) {
    int n = idx - AK_SZ - BN_SZ;
    out[XNEW_SZ + AK_SZ + BN_SZ + n] = (keptInv[n] >= 0) ? 1.0f : 0.0f;
  }
}

// ---------------------------------------------------------------------------
extern "C" void kernel_launch(void* const* d_in, const int* in_sizes, int n_in,
                              void* d_out, int out_size, void* d_ws, size_t ws_size,
                              hipStream_t stream) {
  const float* x = (const float*)d_in[0];
  const int* ei = (const int*)d_in[1];
  const int* ei0 = ei;
  const int* ei1 = ei + 2 * EHALF_;   // edge_index row 1 (shape [2, 131072])
  // d_in[2] = target_edge_count (compile-time 128), d_in[3] = batch (implied)
  float* out = (float*)d_out;

  // workspace carve-up (dwords): mag | order | kept | keptInv | root | coeff | Afin
  float* ws = (float*)d_ws;
  float* w_mag = ws;                               // 8192
  int* w_order = (int*)(ws + 8192);                // 8192
  int* w_kept = (int*)(ws + 16384);                // 2048
  int* w_keptInv = (int*)(ws + 18432);             // 8192
  int* w_root = (int*)(ws + 26624);                // 8192
  float* w_coeff = ws + 34816;                     // 8192
  unsigned* w_A = (unsigned*)(ws + 43008);         // 131072 (G*M*16)

  k_mag<<<(N_ * 32) / 256, 256, 0, stream>>>(x, w_mag);
  k_sort<<<G_, M_, 0, stream>>>(w_mag, w_order, w_kept, w_keptInv);
  k_contract<<<G_, M_, 0, stream>>>(ei0, ei1, w_mag, w_order, w_A, w_root, w_coeff);
  k_gemm<<<G_ * 8, 256, 0, stream>>>(x, w_root, w_coeff, w_kept, out);
  k_outputs<<<(AK_SZ + BN_SZ + N_ + 255) / 256, 256, 0, stream>>>(w_A, w_kept, w_keptInv, out);
}